// NNAttentionHead_13073880449825
// MI455X (gfx1250) — compile-verified
//
#include <hip/hip_runtime.h>
#include <hip/hip_bf16.h>
#include <stdint.h>

// B=4, T=512, C=128, HS=64.  M = B*T = 2048.
// ws layout (floats): aq[2048*128] | ak[2048*128] | v[2048*64]  = 2.62 MB

typedef float v2f_t __attribute__((ext_vector_type(2)));
typedef float v8f_t __attribute__((ext_vector_type(8)));

#define Bn   4
#define Tn   512
#define Cn   128
#define HSn  64
#define Mn   (Bn * Tn)   // 2048

// ---- CDNA5 async global->LDS (ASYNCcnt-tracked), per CDNA5 ISA
__device__ __forceinline__ void async_load_b128(void* lds_ptr, const void* gptr) {
    unsigned lds_off = (unsigned)(uintptr_t)lds_ptr;   // LDS aperture: low 32 bits
    uint64_t ga = (uint64_t)(uintptr_t)gptr;
    asm volatile("global_load_async_to_lds_b128 %0, %1, off"
                 :: "v"(lds_off), "v"(ga) : "memory");
}
__device__ __forceinline__ void wait_async0() {
    asm volatile("s_wait_asynccnt 0x0" ::: "memory");
}

// ---------------------------------------------------------------------------
// Kernel 1: projection GEMMs.  One wave per 16x16 tile, f32 WMMA, K fully
// unrolled so every load is an immediate offset off a loop-invariant base.
// ---------------------------------------------------------------------------
template<bool ADDPOS, int N>
__global__ __launch_bounds__(32) void gemm_kernel(
    const float* __restrict__ x, const float* __restrict__ pos,
    const float* __restrict__ W, float* __restrict__ outp)
{
    constexpr int NT = N / 16;
    int tile = blockIdx.x;
    int mt = tile / NT, nt = tile % NT;
    int lane  = threadIdx.x & 31;
    int r     = lane & 15;          // A row / B col within tile
    int khalf = (lane >> 4) << 1;   // lanes 0-15 -> K k0,k0+1 ; 16-31 -> k0+2,k0+3
    int m0 = mt * 16, n0 = nt * 16;

    const float* xr = x   + (size_t)(m0 + r) * Cn + khalf;
    const float* pr = pos + (size_t)((m0 + r) & (Tn - 1)) * Cn + khalf;
    const float* Wl = W   + (size_t)khalf * N + n0 + r;

    v8f_t acc = {};
    #pragma unroll
    for (int k0 = 0; k0 < Cn; k0 += 4) {
        float2 av = *(const float2*)(xr + k0);
        if (ADDPOS) {
            float2 pv = *(const float2*)(pr + k0);
            av.x += pv.x; av.y += pv.y;
        }
        v2f_t a, b;
        a.x = av.x; a.y = av.y;
        b.x = Wl[(size_t)(k0)     * N];
        b.y = Wl[(size_t)(k0 + 1) * N];
        acc = __builtin_amdgcn_wmma_f32_16x16x4_f32(
            false, a, false, b, (short)0, acc, false, false);
    }
    int rbase = (lane < 16) ? 0 : 8;
    #pragma unroll
    for (int rr = 0; rr < 8; ++rr)
        outp[(size_t)(m0 + rbase + rr) * N + n0 + r] = acc[rr];
}

// ---------------------------------------------------------------------------
// Kernel 2: fused relu-MLP scoring + online causal softmax + P@V (f32 WMMA).
// One workgroup (128 thr = 4 waves) per (batch, 16-row i-tile).  Tile staging
// via global_load_async_to_lds_b128 (ASYNCcnt), fully unrolled issue loops.
// ---------------------------------------------------------------------------
__global__ __launch_bounds__(128) void attn_kernel(
    const float* __restrict__ aq, const float* __restrict__ ak,
    const float* __restrict__ vbuf, const float* __restrict__ b1,
    const float* __restrict__ W2, const float* __restrict__ b2p,
    float* __restrict__ out)
{
    __shared__ float aq_s[16][Cn];       // float4 reads, row-parallel by c
    __shared__ float ak_s[16][132];      // pad keeps b128 alignment (528B rows)
    __shared__ float v_s [16][68];       // pad: both lane halves distinct banks
    __shared__ float p_s [16][17];       // pad: A-frag reads conflict-free
    __shared__ float b1_s[Cn], w2_s[Cn];
    __shared__ float m_s[16], l_s[16], sc_s[16];

    int tid  = threadIdx.x;
    int lane = tid & 31;
    int wave = tid >> 5;
    int b    = blockIdx.x >> 5;          // 32 i-tiles per batch
    int it   = blockIdx.x & 31;
    int i0   = it * 16;
    size_t rowbase = (size_t)b * Tn;

    // per-thread staging coordinates (uniform constant trip counts)
    int rrA = tid >> 5, ccA = (tid & 31) << 2;   // 16x128 tiles: 4 issues/thread
    int rrV = tid >> 4, ccV = (tid & 15) << 2;   // 16x64 tile:   2 issues/thread

    // stage Aq tile asynchronously + small params
    #pragma unroll
    for (int u = 0; u < 4; ++u)
        async_load_b128(&aq_s[rrA + 4 * u][ccA],
                        aq + ((rowbase + i0 + rrA + 4 * u) * Cn + ccA));
    if (tid < Cn) { b1_s[tid] = b1[tid]; w2_s[tid] = W2[tid]; }
    if (tid < 16) { m_s[tid] = -__builtin_inff(); l_s[tid] = 0.f; sc_s[tid] = 0.f; }
    float b2v = b2p[0];
    wait_async0();
    __syncthreads();

    v8f_t acc = {};
    int g = tid >> 4;                    // row group 0..7 (16 lanes each)
    int j = tid & 15;                    // col within j-tile
    int rr16  = lane & 15;
    int khalf = (lane >> 4) << 1;

    for (int jt = 0; jt <= it; ++jt) {
        int j0 = jt * 16;
        __syncthreads();                 // previous tile fully consumed
        #pragma unroll
        for (int u = 0; u < 4; ++u)      // Ak tile -> LDS (async)
            async_load_b128(&ak_s[rrA + 4 * u][ccA],
                            ak + ((rowbase + j0 + rrA + 4 * u) * Cn + ccA));
        #pragma unroll
        for (int u = 0; u < 2; ++u)      // V tile -> LDS (async)
            async_load_b128(&v_s[rrV + 8 * u][ccV],
                            vbuf + ((rowbase + j0 + rrV + 8 * u) * HSn + ccV));
        if (jt < it)                     // next Ak tile -> global_prefetch_b8
            __builtin_prefetch(ak + (rowbase + j0 + 16) * Cn + tid * 4, 0, 1);
        wait_async0();
        __syncthreads();

        // ---- score both rows (g, g+8) vs col j over 128 channels, float4 ----
        const float4* aq0p = (const float4*)&aq_s[g][0];
        const float4* aq1p = (const float4*)&aq_s[g + 8][0];
        const float4* akp  = (const float4*)&ak_s[j][0];
        const float4* b1p  = (const float4*)b1_s;
        const float4* w2p  = (const float4*)w2_s;
        float d0 = 0.f, d1 = 0.f;
        #pragma unroll 4
        for (int cc = 0; cc < Cn / 4; ++cc) {
            float4 ka = akp[cc], bb = b1p[cc], ww = w2p[cc];
            float4 q0 = aq0p[cc], q1 = aq1p[cc];
            float s0 = ka.x + bb.x, s1 = ka.y + bb.y;
            float s2 = ka.z + bb.z, s3 = ka.w + bb.w;
            d0 = fmaf(fmaxf(q0.x + s0, 0.f), ww.x, d0);
            d0 = fmaf(fmaxf(q0.y + s1, 0.f), ww.y, d0);
            d0 = fmaf(fmaxf(q0.z + s2, 0.f), ww.z, d0);
            d0 = fmaf(fmaxf(q0.w + s3, 0.f), ww.w, d0);
            d1 = fmaf(fmaxf(q1.x + s0, 0.f), ww.x, d1);
            d1 = fmaf(fmaxf(q1.y + s1, 0.f), ww.y, d1);
            d1 = fmaf(fmaxf(q1.z + s2, 0.f), ww.z, d1);
            d1 = fmaf(fmaxf(q1.w + s3, 0.f), ww.w, d1);
        }

        // ---- online softmax per row (width-16 reductions, wave32) ----
        #pragma unroll
        for (int pass = 0; pass < 2; ++pass) {
            int i = pass ? (g + 8) : g;
            float dot = pass ? d1 : d0;
            float wei = (dot + b2v) * 0.08838834764831845f;   // C^-0.5
            if (j0 + j > i0 + i) wei = -__builtin_inff();     // causal mask
            float wm = wei;
            #pragma unroll
            for (int off = 8; off; off >>= 1)
                wm = fmaxf(wm, __shfl_xor(wm, off, 16));
            float mold = m_s[i];
            float mnew = fmaxf(mold, wm);
            float p  = __expf(wei - mnew);
            float rs = p;
            #pragma unroll
            for (int off = 8; off; off >>= 1)
                rs += __shfl_xor(rs, off, 16);
            float sc = __expf(mold - mnew);
            p_s[i][j] = p;
            if (j == 0) { m_s[i] = mnew; l_s[i] = l_s[i] * sc + rs; sc_s[i] = sc; }
        }
        __syncthreads();

        // ---- rescale accumulators, then P(16x16) @ V-slice via f32 WMMA ----
        #pragma unroll
        for (int r8 = 0; r8 < 8; ++r8) {
            int row = (lane < 16) ? r8 : r8 + 8;
            acc[r8] *= sc_s[row];
        }
        int n0 = wave * 16;              // each wave owns 16 of the 64 h-cols
        #pragma unroll
        for (int k0 = 0; k0 < 16; k0 += 4) {
            v2f_t a, bf;
            a.x  = p_s[rr16][k0 + khalf];
            a.y  = p_s[rr16][k0 + khalf + 1];
            bf.x = v_s[k0 + khalf    ][n0 + rr16];
            bf.y = v_s[k0 + khalf + 1][n0 + rr16];
            acc = __builtin_amdgcn_wmma_f32_16x16x4_f32(
                false, a, false, bf, (short)0, acc, false, false);
        }
    }
    __syncthreads();

    #pragma unroll
    for (int r8 = 0; r8 < 8; ++r8) {
        int row = (lane < 16) ? r8 : r8 + 8;
        float inv = 1.0f / l_s[row];
        out[(rowbase + i0 + row) * HSn + wave * 16 + rr16] = acc[r8] * inv;
    }
}

// ---------------------------------------------------------------------------
extern "C" void kernel_launch(void* const* d_in, const int* in_sizes, int n_in,
                              void* d_out, int out_size, void* d_ws, size_t ws_size,
                              hipStream_t stream) {
    const float* x   = (const float*)d_in[0];
    const float* pos = (const float*)d_in[1];
    const float* W1  = (const float*)d_in[2];
    const float* b1  = (const float*)d_in[3];
    const float* W2  = (const float*)d_in[4];
    const float* b2  = (const float*)d_in[5];
    const float* Wv  = (const float*)d_in[6];
    float* out = (float*)d_out;

    float* aq  = (float*)d_ws;
    float* ak  = aq + (size_t)Mn * Cn;
    float* vws = ak + (size_t)Mn * Cn;

    gemm_kernel<true, Cn>  <<<(Mn / 16) * (Cn / 16),  32, 0, stream>>>(x, pos, W1 + Cn * Cn, aq);
    gemm_kernel<true, Cn>  <<<(Mn / 16) * (Cn / 16),  32, 0, stream>>>(x, pos, W1,           ak);
    gemm_kernel<false, HSn><<<(Mn / 16) * (HSn / 16), 32, 0, stream>>>(x, pos, Wv,           vws);
    attn_kernel<<<Bn * (Tn / 16), 128, 0, stream>>>(aq, ak, vws, b1, W2, b2, out);
}